// CrossLingualOTModel_9448928051791
// MI455X (gfx1250) — compile-verified
//
#include <hip/hip_runtime.h>
#include <hip/hip_bf16.h>

typedef __attribute__((ext_vector_type(2))) float v2f;
typedef __attribute__((ext_vector_type(8))) float v8f;

#define DEV __device__ __forceinline__

constexpr int B_ = 32, L_ = 384, H_ = 768, KN = 160, GH = 512, GO = 256;
constexpr float ALPHA_ = 0.5f, EPS_ = 0.01f, LRELU_ = 0.2f;
constexpr int NITER_ = 30;

// ---- d_out layout (floats), reference return order: gamma, en_emb, vi_emb, D_en, D_vi, M
constexpr size_t OFF_GAMMA = 0;
constexpr size_t OFF_EEMB  = (size_t)B_ * KN * KN;            // 819200
constexpr size_t OFF_VEMB  = OFF_EEMB + (size_t)B_ * KN * GO; // 2129920
constexpr size_t OFF_DEN   = OFF_VEMB + (size_t)B_ * KN * GO; // 3440640
constexpr size_t OFF_DVI   = OFF_DEN + (size_t)B_ * KN * KN;  // 4259840
constexpr size_t OFF_M     = OFF_DVI + (size_t)B_ * KN * KN;  // 5079040

// ---- workspace layout (floats), ~52 MB total
constexpr size_t WS_Z1  = 0;                                  // B*2*KN*GH (reused for z2)
constexpr size_t WS_H   = WS_Z1 + (size_t)B_ * 2 * KN * GH;   // B*2*KN*GH (reused for normalized emb)
constexpr size_t WS_LG  = WS_H  + (size_t)B_ * 2 * KN * GH;   // B*KN*KN   (ATT overlaps LG+TMP)
constexpr size_t WS_TMP = WS_LG + (size_t)B_ * KN * KN;       // B*KN*KN
constexpr size_t WS_R   = WS_TMP + (size_t)B_ * KN * KN;      // B*2*KN
constexpr size_t WS_G   = WS_R + (size_t)B_ * 2 * KN;         // B*KN*KN  (G = exp(lg))

// One f32 WMMA step: D(16x16) += A(16x4) * B(4x16), fp32 throughout.
DEV v8f wmma4(v2f a, v2f b, v8f c) {
  return __builtin_amdgcn_wmma_f32_16x16x4_f32(
      /*neg_a=*/false, a, /*neg_b=*/false, b,
      /*c_mod=*/(short)0, c, /*reuse_a=*/false, /*reuse_b=*/false);
}

// ============================================================================
// GEMM: C(KN x N) = A(KN x K) @ W(K x N), A either gathered rows of hidden or
// a plain ws matrix. grid.x = B*2 (sample x graph). 8 waves / block.
// ============================================================================
__global__ void gemm_aw(const float* __restrict__ en_h, const float* __restrict__ vi_h,
                        const int* __restrict__ en_k, const int* __restrict__ vi_k,
                        const float* __restrict__ Aplain,
                        const float* __restrict__ W, float* __restrict__ Cout,
                        int K, int N, int gather) {
  const int sg = blockIdx.x, s = sg >> 1, g = sg & 1;
  const int wave = threadIdx.x >> 5, nw = blockDim.x >> 5;
  const int lane = threadIdx.x & 31, half = lane >> 4, lr = lane & 15;
  const int Nt = N / 16, tiles = (KN / 16) * Nt;
  float* o = Cout + (size_t)sg * KN * N;
  for (int t = wave; t < tiles; t += nw) {
    const int m0 = (t / Nt) * 16, n0 = (t - (t / Nt) * Nt) * 16;
    const float* abase;
    if (gather) {
      const int* keep = g ? vi_k : en_k;
      const float* hid = g ? vi_h : en_h;
      abase = hid + (size_t)s * L_ * H_ + (size_t)keep[s * KN + m0 + lr] * H_;
    } else {
      abase = Aplain + (size_t)sg * KN * K + (size_t)(m0 + lr) * K;
    }
    v8f acc = {};
    for (int k0 = 0; k0 < K; k0 += 16) {
      // prefetch upcoming weight rows into the cache hierarchy
      if (k0 + 16 < K)
        __builtin_prefetch(W + (size_t)(k0 + 16 + 2 * half) * N + n0 + lr, 0, 1);
#pragma unroll
      for (int kk = 0; kk < 16; kk += 4) {
        const int k = k0 + kk;
        v2f a = *(const v2f*)(abase + k + 2 * half);
        const float* bp = W + (size_t)(k + 2 * half) * N + n0 + lr;
        v2f b; b.x = bp[0]; b.y = bp[N];
        acc = wmma4(a, b, acc);
      }
    }
    float* ob = o + (size_t)(m0 + 8 * half) * N + n0 + lr;
#pragma unroll
    for (int v = 0; v < 8; ++v) ob[(size_t)v * N] = acc[v];
  }
}

// ============================================================================
// GAT attention scores + masked row softmax. One thread per node row.
// att(i,j) = softmax_j( mask(adj+I, leakyrelu(s_i + d_j)) )
// ============================================================================
__global__ void attn_soft(const float* __restrict__ Z,
                          const float* __restrict__ en_adj, const float* __restrict__ vi_adj,
                          const float* __restrict__ asrc, const float* __restrict__ adst,
                          float* __restrict__ ATT, int N) {
  __shared__ float ss[KN], dd[KN];
  const int sg = blockIdx.x, s = sg >> 1, g = sg & 1;
  const float* z = Z + (size_t)sg * KN * N;
  const float* adj = (g ? vi_adj : en_adj) + (size_t)s * KN * KN;
  float* att = ATT + (size_t)sg * KN * KN;
  const int t = threadIdx.x;
  if (t < KN) {
    float sv = 0.f, dv = 0.f;
    const float* zr = z + (size_t)t * N;
    for (int k = 0; k < N; ++k) { float zv = zr[k]; sv += zv * asrc[k]; dv += zv * adst[k]; }
    ss[t] = sv; dd[t] = dv;
  }
  __syncthreads();
  if (t < KN) {
    float* ar = att + (size_t)t * KN;
    const float* adr = adj + (size_t)t * KN;
    const float si = ss[t];
    float mx = -1e30f;
    for (int j = 0; j < KN; ++j) {
      float e = si + dd[j];
      e = (e > 0.f) ? e : LRELU_ * e;
      e = ((adr[j] > 0.f) || (j == t)) ? e : -1e9f;  // adj with self loops
      ar[j] = e;
      mx = fmaxf(mx, e);
    }
    float sum = 0.f;
    for (int j = 0; j < KN; ++j) { float p = __expf(ar[j] - mx); ar[j] = p; sum += p; }
    const float inv = 1.f / sum;
    for (int j = 0; j < KN; ++j) ar[j] *= inv;
  }
}

// ============================================================================
// out = act( att(KN x KN) @ z(KN x N) ), optional ELU. Output split per graph.
// ============================================================================
__global__ void attn_av(const float* __restrict__ ATT, const float* __restrict__ Z,
                        float* __restrict__ out_en, float* __restrict__ out_vi,
                        size_t s_stride, int N, int act) {
  const int sg = blockIdx.x, s = sg >> 1, g = sg & 1;
  const int wave = threadIdx.x >> 5, nw = blockDim.x >> 5;
  const int lane = threadIdx.x & 31, half = lane >> 4, lr = lane & 15;
  const float* att = ATT + (size_t)sg * KN * KN;
  const float* z = Z + (size_t)sg * KN * N;
  float* o = (g ? out_vi : out_en) + (size_t)s * s_stride;
  const int Nt = N / 16, tiles = (KN / 16) * Nt;
  for (int t = wave; t < tiles; t += nw) {
    const int m0 = (t / Nt) * 16, n0 = (t - (t / Nt) * Nt) * 16;
    const float* ab = att + (size_t)(m0 + lr) * KN;
    v8f acc = {};
    for (int k0 = 0; k0 < KN; k0 += 4) {
      v2f a = *(const v2f*)(ab + k0 + 2 * half);
      const float* bp = z + (size_t)(k0 + 2 * half) * N + n0 + lr;
      v2f b; b.x = bp[0]; b.y = bp[N];
      acc = wmma4(a, b, acc);
    }
    float* ob = o + (size_t)(m0 + 8 * half) * N + n0 + lr;
#pragma unroll
    for (int v = 0; v < 8; ++v) {
      float x = acc[v];
      if (act) x = (x > 0.f) ? x : (__expf(x) - 1.f);  // ELU
      ob[(size_t)v * N] = x;
    }
  }
}

// ============================================================================
// Row-normalize emb (from d_out) into ws.
// ============================================================================
__global__ void normalize_k(const float* __restrict__ out, float* __restrict__ Nb) {
  const int sg = blockIdx.x, s = sg >> 1, g = sg & 1;
  const int t = threadIdx.x;
  if (t >= KN) return;
  const float* e = out + (g ? OFF_VEMB : OFF_EEMB) + (size_t)s * KN * GO + (size_t)t * GO;
  float* n = Nb + (size_t)sg * KN * GO + (size_t)t * GO;
  float sum = 0.f;
  for (int k = 0; k < GO; ++k) sum += e[k] * e[k];
  const float inv = 1.f / fmaxf(sqrtf(sum), 1e-12f);
  for (int k = 0; k < GO; ++k) n[k] = e[k] * inv;
}

// ============================================================================
// P = 1 - X @ Y^T (160x160, K=256). which: 0 -> D_en, 1 -> D_vi, 2 -> M.
// ============================================================================
__global__ void cosdist_k(const float* __restrict__ Nb, float* __restrict__ out) {
  const int idx = blockIdx.x, s = idx / 3, w = idx - 3 * (idx / 3);
  const float* X = Nb + (size_t)(2 * s + (w == 1 ? 1 : 0)) * KN * GO;
  const float* Y = Nb + (size_t)(2 * s + (w == 0 ? 0 : 1)) * KN * GO;
  float* o = out + (w == 0 ? OFF_DEN : (w == 1 ? OFF_DVI : OFF_M)) + (size_t)s * KN * KN;
  const int wave = threadIdx.x >> 5, nw = blockDim.x >> 5;
  const int lane = threadIdx.x & 31, half = lane >> 4, lr = lane & 15;
  const int Nt = KN / 16, tiles = Nt * Nt;
  for (int t = wave; t < tiles; t += nw) {
    const int m0 = (t / Nt) * 16, n0 = (t - (t / Nt) * Nt) * 16;
    const float* ab = X + (size_t)(m0 + lr) * GO;
    const float* bb = Y + (size_t)(n0 + lr) * GO;  // B = Y^T: lane holds column j
    v8f acc = {};
    for (int k0 = 0; k0 < GO; k0 += 4) {
      v2f a = *(const v2f*)(ab + k0 + 2 * half);
      v2f b = *(const v2f*)(bb + k0 + 2 * half);
      acc = wmma4(a, b, acc);
    }
    float* ob = o + (size_t)(m0 + 8 * half) * KN + n0 + lr;
#pragma unroll
    for (int v = 0; v < 8; ++v) ob[(size_t)v * KN] = 1.f - acc[v];
  }
}

// ============================================================================
// FGW init: r1/r2 = (D^2 @ a) with a = 1/KN; lg0 = -2 log KN; G0 = exp(lg0).
// ============================================================================
__global__ void fgw_init(const float* __restrict__ out, float* __restrict__ LG,
                         float* __restrict__ G, float* __restrict__ R) {
  const int s = blockIdx.x, t = threadIdx.x;
  if (t < KN) {
    const float* d1 = out + OFF_DEN + (size_t)s * KN * KN + (size_t)t * KN;
    const float* d2 = out + OFF_DVI + (size_t)s * KN * KN + (size_t)t * KN;
    float r1 = 0.f, r2 = 0.f;
    for (int j = 0; j < KN; ++j) { r1 += d1[j] * d1[j]; r2 += d2[j] * d2[j]; }
    R[(size_t)s * 2 * KN + t] = r1 * (1.0f / KN);
    R[(size_t)s * 2 * KN + KN + t] = r2 * (1.0f / KN);
  }
  const float lg0 = -2.0f * __logf((float)KN);
  const float g0 = 1.0f / ((float)KN * (float)KN);
  float* lg = LG + (size_t)s * KN * KN;
  float* gg = G + (size_t)s * KN * KN;
  for (int i = t; i < KN * KN; i += blockDim.x) { lg[i] = lg0; gg[i] = g0; }
}

// ============================================================================
// FGW matmul 1: tmp = G @ D2^T  (G = exp(lg) maintained by lse/init kernels)
// ============================================================================
__global__ void fgw_mm1(const float* __restrict__ G, const float* __restrict__ out,
                        float* __restrict__ TMP) {
  const int s = blockIdx.x;
  const int wave = threadIdx.x >> 5, nw = blockDim.x >> 5;
  const int lane = threadIdx.x & 31, half = lane >> 4, lr = lane & 15;
  const float* gg = G + (size_t)s * KN * KN;
  const float* D2 = out + OFF_DVI + (size_t)s * KN * KN;
  float* tmp = TMP + (size_t)s * KN * KN;
  const int Nt = KN / 16, tiles = Nt * Nt;
  for (int t = wave; t < tiles; t += nw) {
    const int m0 = (t / Nt) * 16, n0 = (t - (t / Nt) * Nt) * 16;
    const float* ab = gg + (size_t)(m0 + lr) * KN;
    const float* bb = D2 + (size_t)(n0 + lr) * KN;  // D2^T
    v8f acc = {};
    for (int k0 = 0; k0 < KN; k0 += 4) {
      v2f a = *(const v2f*)(ab + k0 + 2 * half);
      v2f b = *(const v2f*)(bb + k0 + 2 * half);
      acc = wmma4(a, b, acc);
    }
    float* ob = tmp + (size_t)(m0 + 8 * half) * KN + n0 + lr;
#pragma unroll
    for (int v = 0; v < 8; ++v) ob[(size_t)v * KN] = acc[v];
  }
}

// ============================================================================
// FGW matmul 2 + update: T = D1 @ tmp;
// lg -= (ALPHA*(r1_i + r2_j - 2T) + (1-ALPHA)*M) / EPS   (fused into D-store)
// ============================================================================
__global__ void fgw_mm2(float* __restrict__ LG, const float* __restrict__ TMP,
                        const float* __restrict__ out, const float* __restrict__ R) {
  const int s = blockIdx.x;
  const int wave = threadIdx.x >> 5, nw = blockDim.x >> 5;
  const int lane = threadIdx.x & 31, half = lane >> 4, lr = lane & 15;
  const float* D1 = out + OFF_DEN + (size_t)s * KN * KN;
  const float* Mm = out + OFF_M + (size_t)s * KN * KN;
  const float* tmp = TMP + (size_t)s * KN * KN;
  const float* r1 = R + (size_t)s * 2 * KN;
  const float* r2 = r1 + KN;
  float* lg = LG + (size_t)s * KN * KN;
  const int Nt = KN / 16, tiles = Nt * Nt;
  for (int t = wave; t < tiles; t += nw) {
    const int m0 = (t / Nt) * 16, n0 = (t - (t / Nt) * Nt) * 16;
    const float* ab = D1 + (size_t)(m0 + lr) * KN;
    v8f acc = {};
    for (int k0 = 0; k0 < KN; k0 += 4) {
      v2f a = *(const v2f*)(ab + k0 + 2 * half);
      const float* bp = tmp + (size_t)(k0 + 2 * half) * KN + n0 + lr;
      v2f b; b.x = bp[0]; b.y = bp[KN];
      acc = wmma4(a, b, acc);
    }
    const int j = n0 + lr;
#pragma unroll
    for (int v = 0; v < 8; ++v) {
      const int i = m0 + 8 * half + v;
      const float c1 = r1[i] + r2[j];
      const float grad = ALPHA_ * (c1 - 2.f * acc[v]) + (1.f - ALPHA_) * Mm[(size_t)i * KN + j];
      lg[(size_t)i * KN + j] -= grad * (1.f / EPS_);
    }
  }
}

// ============================================================================
// Log-domain marginal projections: lg += (la - logsumexp(lg, axis)) broadcast.
// Also refresh G = exp(lg) for the next mm1 (single traversal, no redundancy).
// ============================================================================
__global__ void fgw_lse_row(float* __restrict__ LG, float* __restrict__ G) {
  const int s = blockIdx.x, t = threadIdx.x;
  if (t >= KN) return;
  float* r = LG + (size_t)s * KN * KN + (size_t)t * KN;
  float* gr = G + (size_t)s * KN * KN + (size_t)t * KN;
  float mx = -1e30f;
  for (int j = 0; j < KN; ++j) mx = fmaxf(mx, r[j]);
  float sum = 0.f;
  for (int j = 0; j < KN; ++j) sum += __expf(r[j] - mx);
  const float la = -__logf((float)KN);
  const float adj = la - (mx + __logf(sum));
  for (int j = 0; j < KN; ++j) {
    const float x = r[j] + adj;
    r[j] = x;
    gr[j] = __expf(x);
  }
}

__global__ void fgw_lse_col(float* __restrict__ LG, float* __restrict__ G) {
  const int s = blockIdx.x, t = threadIdx.x;
  if (t >= KN) return;
  float* c = LG + (size_t)s * KN * KN + t;
  float* gc = G + (size_t)s * KN * KN + t;
  float mx = -1e30f;
  for (int j = 0; j < KN; ++j) mx = fmaxf(mx, c[(size_t)j * KN]);
  float sum = 0.f;
  for (int j = 0; j < KN; ++j) sum += __expf(c[(size_t)j * KN] - mx);
  const float la = -__logf((float)KN);
  const float adj = la - (mx + __logf(sum));
  for (int j = 0; j < KN; ++j) {
    const float x = c[(size_t)j * KN] + adj;
    c[(size_t)j * KN] = x;
    gc[(size_t)j * KN] = __expf(x);
  }
}

// ============================================================================
extern "C" void kernel_launch(void* const* d_in, const int* in_sizes, int n_in,
                              void* d_out, int out_size, void* d_ws, size_t ws_size,
                              hipStream_t stream) {
  (void)in_sizes; (void)n_in; (void)out_size; (void)ws_size;
  const float* en_h  = (const float*)d_in[0];
  const float* vi_h  = (const float*)d_in[1];
  const int*   en_k  = (const int*)d_in[2];
  const int*   vi_k  = (const int*)d_in[3];
  const float* en_adj = (const float*)d_in[4];
  const float* vi_adj = (const float*)d_in[5];
  const float* W1  = (const float*)d_in[6];
  const float* a1s = (const float*)d_in[7];
  const float* a1d = (const float*)d_in[8];
  const float* W2  = (const float*)d_in[9];
  const float* a2s = (const float*)d_in[10];
  const float* a2d = (const float*)d_in[11];
  float* out = (float*)d_out;
  float* ws  = (float*)d_ws;

  float* Z1  = ws + WS_Z1;   // also z2
  float* Hb  = ws + WS_H;    // also normalized emb
  float* LG  = ws + WS_LG;
  float* TMP = ws + WS_TMP;
  float* R   = ws + WS_R;
  float* G   = ws + WS_G;
  float* ATT = ws + WS_LG;   // overlaps LG+TMP (disjoint in time)

  // GAT layer 1: z1 = gather(hidden) @ W1; h = elu(att1 @ z1)
  gemm_aw<<<B_ * 2, 256, 0, stream>>>(en_h, vi_h, en_k, vi_k, nullptr, W1, Z1, H_, GH, 1);
  attn_soft<<<B_ * 2, 192, 0, stream>>>(Z1, en_adj, vi_adj, a1s, a1d, ATT, GH);
  attn_av<<<B_ * 2, 256, 0, stream>>>(ATT, Z1, Hb, Hb + (size_t)KN * GH,
                                      (size_t)2 * KN * GH, GH, 1);
  // GAT layer 2: z2 = h @ W2; emb = att2 @ z2  (emb straight to d_out)
  gemm_aw<<<B_ * 2, 256, 0, stream>>>(nullptr, nullptr, nullptr, nullptr, Hb, W2, Z1, GH, GO, 0);
  attn_soft<<<B_ * 2, 192, 0, stream>>>(Z1, en_adj, vi_adj, a2s, a2d, ATT, GO);
  attn_av<<<B_ * 2, 256, 0, stream>>>(ATT, Z1, out + OFF_EEMB, out + OFF_VEMB,
                                      (size_t)KN * GO, GO, 0);
  // Normalize + structure/cost matrices D_en, D_vi, M
  normalize_k<<<B_ * 2, 192, 0, stream>>>(out, Hb);
  cosdist_k<<<B_ * 3, 256, 0, stream>>>(Hb, out);

  // Entropic FGW, Bregman alternating projections (log domain)
  fgw_init<<<B_, 256, 0, stream>>>(out, LG, G, R);
  for (int it = 0; it < NITER_; ++it) {
    fgw_mm1<<<B_, 256, 0, stream>>>(G, out, TMP);
    fgw_mm2<<<B_, 256, 0, stream>>>(LG, TMP, out, R);
    fgw_lse_row<<<B_, 192, 0, stream>>>(LG, G);
    fgw_mm1<<<B_, 256, 0, stream>>>(G, out, TMP);
    fgw_mm2<<<B_, 256, 0, stream>>>(LG, TMP, out, R);
    fgw_lse_col<<<B_, 192, 0, stream>>>(LG, G);
  }
  // gamma = exp(lg_final) == G written by the last fgw_lse_col
  hipMemcpyAsync(out + OFF_GAMMA, G, (size_t)B_ * KN * KN * sizeof(float),
                 hipMemcpyDeviceToDevice, stream);
}